// ConcatCritic_11922829214038
// MI455X (gfx1250) — compile-verified
//
#include <hip/hip_runtime.h>

#define B_  512
#define DX_ 128
#define DY_ 128
#define H_  256
#define LDH 260   // padded LDS row stride (floats): l16*260 + 2*lhi hits distinct banks

typedef __attribute__((ext_vector_type(2))) float v2f;
typedef __attribute__((ext_vector_type(8))) float v8f;

// ---------------- Stage 1: hx = x @ W1[:DX], hy = y @ W1[DX:] + b1 ----------------
__global__ __launch_bounds__(H_) void hxhy_kernel(
    const float* __restrict__ x, const float* __restrict__ y,
    const float* __restrict__ W1, const float* __restrict__ b1,
    float* __restrict__ hx, float* __restrict__ hy) {
  __shared__ float xs[DX_];
  __shared__ float ys[DY_];
  const int b = blockIdx.x;
  const int n = threadIdx.x;
  if (n < DX_) xs[n] = x[b * DX_ + n];
  else         ys[n - DX_] = y[b * DY_ + (n - DX_)];
  __syncthreads();
  float ax = 0.f, ay = 0.f;
#pragma unroll 8
  for (int k = 0; k < DX_; ++k) {
    ax = fmaf(xs[k], W1[k * H_ + n], ax);
    ay = fmaf(ys[k], W1[(DX_ + k) * H_ + n], ay);
  }
  hx[b * H_ + n] = ax;
  hy[b * H_ + n] = ay + b1[n];
}

// ---------------- Stage 2: fused relu -> GEMM(W2) -> relu -> dot(W3) ----------------
// One workgroup (256 thr = 8 waves) computes a 16x16 tile of scores.
// M-dimension = 256 pairs; wave w owns pairs [32w, 32w+32) = two 16-row WMMA tiles,
// each tile being (one i row) x (all 16 j's).
template <bool STAGED>
__global__ __launch_bounds__(256, 1) void critic_kernel(
    const float* __restrict__ hx, const float* __restrict__ hy,
    const float* __restrict__ x,  const float* __restrict__ y,
    const float* __restrict__ W1, const float* __restrict__ b1,
    const float* __restrict__ W2, const float* __restrict__ b2,
    const float* __restrict__ W3, const float* __restrict__ b3,
    float* __restrict__ out) {
  __shared__ float hxs[16 * LDH];   // hx rows i0..i0+15
  __shared__ float hys[16 * LDH];   // hy rows j0..j0+15 (b1 already folded in)
  __shared__ float w2s[16 * LDH];   // W2 N-tile, transposed: w2s[n*LDH + k]
  __shared__ float b2s[H_];
  __shared__ float w3s[H_];

  const int t  = threadIdx.x;
  const int i0 = blockIdx.y * 16;
  const int j0 = blockIdx.x * 16;

  b2s[t] = b2[t];
  w3s[t] = W3[t];

  if (STAGED) {
    for (int idx = t; idx < 16 * H_; idx += 256) {
      const int r = idx >> 8, c = idx & (H_ - 1);
      hxs[r * LDH + c] = hx[(i0 + r) * H_ + c];
      hys[r * LDH + c] = hy[(j0 + r) * H_ + c];
    }
    __syncthreads();
  } else {
    // Fallback: compute the hx/hy tiles in-kernel (uses w2s as x/y staging).
    float* xs = w2s;                // 16 x 128
    float* ys = w2s + 16 * DX_;     // 16 x 128
    for (int idx = t; idx < 16 * DX_; idx += 256) {
      const int r = idx >> 7, c = idx & (DX_ - 1);
      xs[idx] = x[(i0 + r) * DX_ + c];
      ys[idx] = y[(j0 + r) * DY_ + c];
    }
    __syncthreads();
    for (int idx = t; idx < 16 * H_; idx += 256) {
      const int r = idx >> 8, c = idx & (H_ - 1);
      float ax = 0.f, ay = 0.f;
#pragma unroll 8
      for (int k = 0; k < DX_; ++k) {
        ax = fmaf(xs[r * DX_ + k], W1[k * H_ + c], ax);
        ay = fmaf(ys[r * DX_ + k], W1[(DX_ + k) * H_ + c], ay);
      }
      hxs[r * LDH + c] = ax;
      hys[r * LDH + c] = ay + b1[c];
    }
    __syncthreads();
  }

  const int wave = t >> 5;
  const int lane = t & 31;
  const int l16  = lane & 15;
  const int lhi  = lane >> 4;

  // A-frag (16x4 f32): lane -> row m=l16, regs cover K = 2*lhi, 2*lhi+1.
  // B-frag mirrors with n=l16. Row of hy/w2s indexed by l16, hx by the tile's i.
  const int arow   = l16 * LDH + 2 * lhi;           // shared by hy and w2s fragments
  const int hx0row = (2 * wave) * LDH + 2 * lhi;    // M-tile 0 -> i row 2*wave
  const int hx1row = (2 * wave + 1) * LDH + 2 * lhi;// M-tile 1 -> i row 2*wave+1

  float sacc0[8], sacc1[8];
#pragma unroll
  for (int r = 0; r < 8; ++r) { sacc0[r] = 0.f; sacc1[r] = 0.f; }

  const float b3v = b3[0];

  for (int nt = 0; nt < 16; ++nt) {
    const int n0 = nt * 16;
    __syncthreads();
    // Stage W2[:, n0:n0+16] transposed into LDS (coalesced global reads).
    for (int idx = t; idx < 16 * H_; idx += 256) {
      const int k = idx >> 4, n = idx & 15;
      w2s[n * LDH + k] = W2[k * H_ + n0 + n];
    }
    __syncthreads();

    const float cinit = b2s[n0 + l16];   // C init = b2 broadcast down column n
    v8f acc0, acc1;
#pragma unroll
    for (int r = 0; r < 8; ++r) { acc0[r] = cinit; acc1[r] = cinit; }

#pragma unroll 4
    for (int k = 0; k < H_; k += 4) {
      const v2f bf  = *(const v2f*)&w2s[arow + k];
      const v2f hyv = *(const v2f*)&hys[arow + k];
      const v2f h0  = *(const v2f*)&hxs[hx0row + k];
      const v2f h1  = *(const v2f*)&hxs[hx1row + k];
      v2f a0, a1;
      a0.x = fmaxf(h0.x + hyv.x, 0.f);
      a0.y = fmaxf(h0.y + hyv.y, 0.f);
      a1.x = fmaxf(h1.x + hyv.x, 0.f);
      a1.y = fmaxf(h1.y + hyv.y, 0.f);
      acc0 = __builtin_amdgcn_wmma_f32_16x16x4_f32(false, a0, false, bf, (short)0,
                                                   acc0, false, false);
      acc1 = __builtin_amdgcn_wmma_f32_16x16x4_f32(false, a1, false, bf, (short)0,
                                                   acc1, false, false);
    }

    const float w3v = w3s[n0 + l16];
#pragma unroll
    for (int r = 0; r < 8; ++r) {
      sacc0[r] += fmaxf(acc0[r], 0.f) * w3v;
      sacc1[r] += fmaxf(acc1[r], 0.f) * w3v;
    }
  }

  // Sum over the 16 N-lanes (C/D layout: reg r, lane l -> M = r + 8*lhi, N = l16).
#pragma unroll
  for (int r = 0; r < 8; ++r) {
    float v0 = sacc0[r], v1 = sacc1[r];
    v0 += __shfl_xor(v0, 8, 32);  v1 += __shfl_xor(v1, 8, 32);
    v0 += __shfl_xor(v0, 4, 32);  v1 += __shfl_xor(v1, 4, 32);
    v0 += __shfl_xor(v0, 2, 32);  v1 += __shfl_xor(v1, 2, 32);
    v0 += __shfl_xor(v0, 1, 32);  v1 += __shfl_xor(v1, 1, 32);
    if (l16 == 0) {
      const int p0 = wave * 32 + r + 8 * lhi;   // pair index in tile (Mtile 0)
      const int p1 = p0 + 16;                   // Mtile 1
      out[(i0 + (p0 >> 4)) * B_ + (j0 + (p0 & 15))] = v0 + b3v;
      out[(i0 + (p1 >> 4)) * B_ + (j0 + (p1 & 15))] = v1 + b3v;
    }
  }
}

extern "C" void kernel_launch(void* const* d_in, const int* in_sizes, int n_in,
                              void* d_out, int out_size, void* d_ws, size_t ws_size,
                              hipStream_t stream) {
  const float* x  = (const float*)d_in[0];
  const float* y  = (const float*)d_in[1];
  const float* W1 = (const float*)d_in[2];
  const float* b1 = (const float*)d_in[3];
  const float* W2 = (const float*)d_in[4];
  const float* b2 = (const float*)d_in[5];
  const float* W3 = (const float*)d_in[6];
  const float* b3 = (const float*)d_in[7];
  float* out = (float*)d_out;

  const size_t need = (size_t)2 * B_ * H_ * sizeof(float);   // 1 MB for hx+hy
  dim3 grid(B_ / 16, B_ / 16);

  if (d_ws != nullptr && ws_size >= need) {
    float* hx = (float*)d_ws;
    float* hy = hx + (size_t)B_ * H_;
    hxhy_kernel<<<B_, H_, 0, stream>>>(x, y, W1, b1, hx, hy);
    critic_kernel<true><<<grid, 256, 0, stream>>>(hx, hy, x, y, W1, b1,
                                                  W2, b2, W3, b3, out);
  } else {
    critic_kernel<false><<<grid, 256, 0, stream>>>(nullptr, nullptr, x, y, W1, b1,
                                                   W2, b2, W3, b3, out);
  }
}